// RGCNEncoder_32341103739367
// MI455X (gfx1250) — compile-verified
//
#include <hip/hip_runtime.h>
#include <hip/hip_bf16.h>

// ---------------------------------------------------------------------------
// RGCN encoder for MI455X (gfx1250, wave32, WMMA + async-to-LDS).
// Math (exact rewrite of the reference recurrence):
//   h[j] = c0[j]*(X[j]@Wself + bself)
//        + sum_s g_s[j]*(agg_s[j]@W_s)
//        + sum_s g_s[j]*count_s[j]*b_s
// with g_s[j] = prod over slot s and later slots (in reference order) of 2/max(count,1).
// ---------------------------------------------------------------------------

#define N_NODES 50000
#define EMB 128
#define N_REL 9
#define S2 (2 * N_REL)     // 18
#define NW (S2 + 1)        // 19 weight slots
#define N_EDGES 800000
#define WSLOT (EMB * EMB)  // 16384 floats per weight slot (packed or not)

typedef __attribute__((ext_vector_type(2))) float v2f;
typedef __attribute__((ext_vector_type(8))) float v8f;

// ---------------------------------------------------------------- counts ----
__global__ void rgcn_count_kernel(const int* __restrict__ ei,
                                  const int* __restrict__ et,
                                  float* __restrict__ counts) {
    int i = blockIdx.x * blockDim.x + threadIdx.x;
    if (i >= 2 * N_EDGES) return;
    int dir = (i >= N_EDGES) ? 1 : 0;
    int e = dir ? (i - N_EDGES) : i;
    int s = et[e] + dir * N_REL;
    int dest = dir ? ei[N_EDGES + e] : ei[e];
    atomicAdd(&counts[(size_t)s * N_NODES + dest], 1.0f);
}

// --------------------------------------------------------- coefficients ----
// Reference slot order: for r in 0..8: for inv in 0,1: s = r + 9*inv
__global__ void rgcn_coeff_kernel(const float* __restrict__ counts,
                                  float* __restrict__ coeff,
                                  float* __restrict__ bscale,
                                  float* __restrict__ c0) {
    int j = blockIdx.x * blockDim.x + threadIdx.x;
    if (j >= N_NODES) return;
    const int ord[S2] = {0, 9, 1, 10, 2, 11, 3, 12, 4, 13,
                         5, 14, 6, 15, 7, 16, 8, 17};
    float g = 1.0f;
    #pragma unroll
    for (int p = S2 - 1; p >= 0; --p) {
        int s = ord[p];
        float craw = counts[(size_t)s * N_NODES + j];
        float cnt = craw > 1.0f ? craw : 1.0f;
        g = g * 2.0f / cnt;
        coeff[(size_t)s * N_NODES + j] = g;
        bscale[(size_t)s * N_NODES + j] = g * craw;
    }
    c0[j] = g;
}

// ------------------------------------------------- weight fragment pack ----
// Repack W[slot][128][128] into fragment-major order so that per (k-step,
// n-tile-pair) each lane's 4 B values are one contiguous float4:
//   Wpack[((slot*32 + kk)*4 + g)*128 + lane*4 + c] =
//       W[slot][kk*4 + 2*(lane>>4) + (c&1)][(2*g + (c>>1))*16 + (lane&15)]
__global__ void rgcn_pack_w_kernel(const float* __restrict__ Wt,
                                   float* __restrict__ Wpack) {
    int i = blockIdx.x * blockDim.x + threadIdx.x;
    if (i >= 2 * NW * WSLOT) return;
    int slot = i >> 14;
    int rem  = i & 16383;
    int kk   = rem >> 9;
    int rem2 = rem & 511;
    int g    = rem2 >> 7;
    int q    = rem2 & 127;
    int lane = q >> 2, c = q & 3;
    int kh = lane >> 4, nl = lane & 15;
    int krow = kk * 4 + 2 * kh + (c & 1);
    int col  = (2 * g + (c >> 1)) * 16 + nl;
    Wpack[i] = Wt[(size_t)slot * WSLOT + krow * EMB + col];
}

// ---------------------------------------------------- bias / H init ---------
__global__ void rgcn_init_hidden_kernel(const float* __restrict__ c0,
                                        const float* __restrict__ bscale,
                                        const float* __restrict__ b, // [19][128]
                                        float* __restrict__ H) {
    int i = blockIdx.x * blockDim.x + threadIdx.x;
    if (i >= N_NODES * EMB) return;
    int j = i >> 7;
    int c = i & 127;
    float acc = c0[j] * b[S2 * EMB + c];
    #pragma unroll
    for (int s = 0; s < S2; ++s)
        acc += bscale[(size_t)s * N_NODES + j] * b[s * EMB + c];
    H[i] = acc;
}

// -------------------------------------------------------------- scatter ----
// One wave per (edge, direction); lane handles 4 floats of the 128-wide row.
__global__ void rgcn_scatter_kernel(const int* __restrict__ ei,
                                    const int* __restrict__ et,
                                    const float* __restrict__ X,
                                    float* __restrict__ agg,
                                    int chunkStart, int chunkSize) {
    int wid = (blockIdx.x * blockDim.x + threadIdx.x) >> 5;
    int lane = threadIdx.x & 31;
    if (wid >= 2 * N_EDGES) return;
    int dir = (wid >= N_EDGES) ? 1 : 0;
    int e = dir ? (wid - N_EDGES) : wid;
    int s = et[e] + dir * N_REL - chunkStart;
    if ((unsigned)s >= (unsigned)chunkSize) return;
    int dest = dir ? ei[N_EDGES + e] : ei[e];
    int src  = dir ? ei[e] : ei[N_EDGES + e];
    const float4 v = ((const float4*)(X + (size_t)src * EMB))[lane];
    float* d = agg + (((size_t)s * N_NODES + dest) << 7) + lane * 4;
    atomicAdd(d + 0, v.x);
    atomicAdd(d + 1, v.y);
    atomicAdd(d + 2, v.z);
    atomicAdd(d + 3, v.w);
}

// ----------------------------------------------------------------- GEMM ----
// H[16-row tile] += sum over items of (scale_row * A) @ W, using
// v_wmma_f32_16x16x4_f32. Block = 320 threads = 10 waves = 5 row tiles x 2
// column halves; 625 blocks cover exactly 3125 row tiles (EXEC all ones).
// A tiles are staged to LDS with global_load_async_to_lds_b128 (ASYNCcnt);
// the per-row coefficient is applied to the A fragment in registers.
__global__ __launch_bounds__(320) void rgcn_gemm_acc_kernel(
    const float* __restrict__ agg,        // [Q][N][128]
    const float* __restrict__ coeffChunk, // [Q][N]
    const float* __restrict__ Wp,         // packed [Q][32][4][128]
    int Q,
    const float* __restrict__ X,          // selfloop input (or nullptr)
    const float* __restrict__ c0,         // [N]
    const float* __restrict__ WpSelf,     // packed [32][4][128]
    float* __restrict__ H) {
    __shared__ float As[5][16 * 132];     // padded rows: conflict-free b64 reads

    const int lane = threadIdx.x & 31;
    const int wv = threadIdx.x >> 5;
    const int pair = wv >> 1;             // row tile within block (0..4)
    const int half = wv & 1;              // column half (0..1)
    const int row0 = (blockIdx.x * 5 + pair) << 4;
    const int m = lane & 15;              // M index (A) / N index (B)
    const int kh = lane >> 4;             // K-half selector

    v8f acc[4] = {};                      // 4 n-tiles of 16x16 f32

    const int items = Q + (X ? 1 : 0);
    for (int it = 0; it < items; ++it) {
        const bool selfloop = (it == Q);
        const float* A  = selfloop ? X  : (agg + (size_t)it * N_NODES * EMB);
        const float* sc = selfloop ? c0 : (coeffChunk + (size_t)it * N_NODES);
        const float4* W4 = (const float4*)(selfloop ? WpSelf
                                                    : (Wp + (size_t)it * WSLOT));

        // Async stage of the 16x128 A tile into LDS (pure copy, row-padded).
        {
            int p = half * 32 + lane;     // 0..63 across the wave pair
            #pragma unroll
            for (int t = 0; t < 8; ++t) {
                int linear = p + 64 * t;  // 0..511 float4s
                int r = linear >> 5;
                int c4 = linear & 31;
                unsigned long long ga = (unsigned long long)(uintptr_t)
                    (A + (size_t)(row0 + r) * EMB + c4 * 4);
                unsigned int la = (unsigned int)(uintptr_t)
                    (&As[pair][r * 132 + c4 * 4]);
                asm volatile("global_load_async_to_lds_b128 %0, %1, off"
                             :: "v"(la), "v"(ga) : "memory");
            }
            asm volatile("s_wait_asynccnt 0x0" ::: "memory");
        }
        __syncthreads();

        const float smul = sc[row0 + m];  // per-row coefficient for this lane
        #pragma unroll 4
        for (int kk = 0; kk < 32; ++kk) {
            // A frag: VGPR0=A[m][4kk+2kh], VGPR1=A[m][4kk+2kh+1]
            v2f a = *(const v2f*)&As[pair][m * 132 + kk * 4 + 2 * kh];
            a.x *= smul;
            a.y *= smul;
            // Two b128 loads give four ready B fragments (even-aligned pairs).
            float4 f0 = W4[(kk * 4 + half * 2 + 0) * 32 + lane];
            float4 f1 = W4[(kk * 4 + half * 2 + 1) * 32 + lane];
            v2f b0 = {f0.x, f0.y};
            v2f b1 = {f0.z, f0.w};
            v2f b2 = {f1.x, f1.y};
            v2f b3 = {f1.z, f1.w};
            acc[0] = __builtin_amdgcn_wmma_f32_16x16x4_f32(
                false, a, false, b0, (short)0, acc[0], false, false);
            acc[1] = __builtin_amdgcn_wmma_f32_16x16x4_f32(
                false, a, false, b1, (short)0, acc[1], false, false);
            acc[2] = __builtin_amdgcn_wmma_f32_16x16x4_f32(
                false, a, false, b2, (short)0, acc[2], false, false);
            acc[3] = __builtin_amdgcn_wmma_f32_16x16x4_f32(
                false, a, false, b3, (short)0, acc[3], false, false);
        }
        __syncthreads();
    }

    // Epilogue: H += acc (each wave owns its 16x64 tile exclusively).
    #pragma unroll
    for (int t = 0; t < 4; ++t) {
        int col = half * 64 + t * 16 + m;
        #pragma unroll
        for (int v = 0; v < 8; ++v) {
            size_t idx = (size_t)(row0 + v + 8 * kh) * EMB + col;
            H[idx] += acc[t][v];
        }
    }
}

// ----------------------------------------------------------------- relu ----
__global__ void rgcn_relu_kernel(float* __restrict__ Xio) {
    int i = blockIdx.x * blockDim.x + threadIdx.x;
    if (i >= N_NODES * EMB) return;
    float v = Xio[i];
    Xio[i] = v > 0.0f ? v : 0.0f;
}

// ---------------------------------------------------------------------------
extern "C" void kernel_launch(void* const* d_in, const int* in_sizes, int n_in,
                              void* d_out, int out_size, void* d_ws, size_t ws_size,
                              hipStream_t stream) {
    const int*   edge_index = (const int*)d_in[0];   // [2][E]
    const int*   edge_type  = (const int*)d_in[1];   // [E]
    const float* emb        = (const float*)d_in[2]; // [N][128]
    const float* Wt         = (const float*)d_in[3]; // [2][19][128][128]
    const float* Bs         = (const float*)d_in[4]; // [2][19][128]
    float* out = (float*)d_out;                      // [N][128]

    char* ws = (char*)d_ws;
    size_t off = 0;
    auto alloc = [&](size_t bytes) -> void* {
        void* p = ws + off;
        off += (bytes + 255) & ~(size_t)255;
        return p;
    };
    float* counts = (float*)alloc((size_t)S2 * N_NODES * 4);
    float* coeff  = (float*)alloc((size_t)S2 * N_NODES * 4);
    float* bscale = (float*)alloc((size_t)S2 * N_NODES * 4);
    float* c0     = (float*)alloc((size_t)N_NODES * 4);
    float* inter  = (float*)alloc((size_t)N_NODES * EMB * 4);
    float* Wpack  = (float*)alloc((size_t)2 * NW * WSLOT * 4);

    const size_t slotBytes = (size_t)N_NODES * EMB * 4; // one agg relation slot
    size_t remain = (ws_size > off) ? (ws_size - off) : 0;
    int G = (int)(remain / slotBytes);   // resident relation slots
    if (G < 1) G = 1;
    if (G > S2) G = S2;
    float* agg = (float*)(ws + off);

    // Counts + per-node coefficients (layer-independent) + weight repack.
    hipMemsetAsync(counts, 0, (size_t)S2 * N_NODES * 4, stream);
    rgcn_count_kernel<<<(2 * N_EDGES + 255) / 256, 256, 0, stream>>>(
        edge_index, edge_type, counts);
    rgcn_coeff_kernel<<<(N_NODES + 255) / 256, 256, 0, stream>>>(
        counts, coeff, bscale, c0);
    rgcn_pack_w_kernel<<<(2 * NW * WSLOT + 255) / 256, 256, 0, stream>>>(
        Wt, Wpack);

    const long long scatterThreads = 2LL * N_EDGES * 32;
    const unsigned scatterGrid = (unsigned)((scatterThreads + 255) / 256);

    for (int l = 0; l < 2; ++l) {
        const float* X = (l == 0) ? emb : inter;
        float* H = (l == 0) ? inter : out;
        const float* WlP = Wpack + (size_t)l * NW * WSLOT;
        const float* Bl  = Bs + (size_t)l * NW * EMB;

        rgcn_init_hidden_kernel<<<(N_NODES * EMB + 255) / 256, 256, 0, stream>>>(
            c0, bscale, Bl, H);

        bool first = true;
        for (int cs = 0; cs < S2; cs += G) {
            int Gsz = (S2 - cs < G) ? (S2 - cs) : G;
            hipMemsetAsync(agg, 0, (size_t)Gsz * slotBytes, stream);
            rgcn_scatter_kernel<<<scatterGrid, 256, 0, stream>>>(
                edge_index, edge_type, X, agg, cs, Gsz);
            rgcn_gemm_acc_kernel<<<625, 320, 0, stream>>>(
                agg, coeff + (size_t)cs * N_NODES, WlP + (size_t)cs * WSLOT,
                Gsz, first ? X : nullptr, c0, WlP + (size_t)S2 * WSLOT, H);
            first = false;
        }
        if (l == 0)
            rgcn_relu_kernel<<<(N_NODES * EMB + 255) / 256, 256, 0, stream>>>(inter);
    }
}